// OneCell_72645076844918
// MI455X (gfx1250) — compile-verified
//
#include <hip/hip_runtime.h>
#include <hip/hip_bf16.h>

typedef unsigned int       u32;
typedef unsigned long long u64;
typedef u32 v4u __attribute__((ext_vector_type(4)));
typedef int  v8i __attribute__((ext_vector_type(8)));
typedef int  v4i __attribute__((ext_vector_type(4)));

// Problem geometry (reference: B=16384, T=1024, hidden=1)
constexpr int kT      = 1024;        // sequence length
constexpr int kTileT  = 128;         // timesteps per TDM tile
constexpr int kTiles  = kT / kTileT;
constexpr int kRows   = 32;          // batch rows per block == blockDim.x (1 wave)
constexpr int kStride = kTileT + 1;  // padded LDS row stride (129 ≡ 1 mod 64 banks)

#if __has_builtin(__builtin_amdgcn_tanhf)
#define FAST_TANH(x) __builtin_amdgcn_tanhf(x)
#elif __has_builtin(__builtin_amdgcn_tanh_f32)
#define FAST_TANH(x) __builtin_amdgcn_tanh_f32(x)
#else
#define FAST_TANH(x) tanhf(x)
#endif

// Byte offset of a __shared__ object inside the workgroup LDS allocation.
__device__ __forceinline__ u32 lds_off(const void* p) {
  return (u32)(u64)(__attribute__((address_space(3))) const char*)p;
}

// TDM load of a [kRows x kTileT] f32 tile (memory row stride kT elements)
// into LDS at lds_addr, padded to kStride floats per LDS row.
__device__ __forceinline__ void tdm_load_tile(u64 gaddr, u32 lds_addr) {
  // ---- D# group 0 (128b): count=1 | lds_addr | global_addr[56:0] | type=2
  v4u g0;
  g0[0] = 1u;                                   // count=1 (valid user descriptor)
  g0[1] = lds_addr;                             // LDS byte address
  g0[2] = (u32)(gaddr & 0xffffffffull);         // global_addr[31:0]
  g0[3] = (u32)((gaddr >> 32) & 0x01ffffffull)  // global_addr[56:32]
          | (2u << 30);                         // type = 2 ("image")
  // ---- D# group 1 (256b)
  v8i g1;
  g1[0] = (int)((2u << 16)      // data_size = 2 -> 4 bytes
              | (1u << 20)      // pad_enable
              | (6u << 22)      // pad_interval = 6 -> every 128 dwords
              | (0u << 25));    // pad_amount   = 0 -> 1 dword => LDS stride 129
  g1[1] = (int)((u32)kTileT << 16);  // tensor_dim0 = tile width  (never OOB)
  g1[2] = (int)((u32)kRows  << 16);  // tensor_dim1 = tile height
  g1[3] = (int)((u32)kTileT << 16);  // tile_dim0
  g1[4] = (int)kRows;                // tile_dim1 (tile_dim2 = 0)
  g1[5] = (int)kT;                   // tensor_dim0_stride (elements)
  g1[6] = 0;
  g1[7] = 0;
  v4i gz = {0, 0, 0, 0};             // groups 2/3 unused (2-D tile)
#if __clang_major__ >= 23
  v8i gz8 = {0, 0, 0, 0, 0, 0, 0, 0};
  __builtin_amdgcn_tensor_load_to_lds(g0, g1, gz, gz, gz8, 0);
#else
  __builtin_amdgcn_tensor_load_to_lds(g0, g1, gz, gz, 0);
#endif
}

__global__ __launch_bounds__(kRows)
void OneCell_lstm_kernel(const float* __restrict__ x,
                         const float* __restrict__ w_ih,
                         const float* __restrict__ w_hh,
                         const float* __restrict__ b_ih,
                         const float* __restrict__ b_hh,
                         float* __restrict__ out) {
  // Single-wave workgroup: no barriers needed; the wave self-paces on
  // TENSORcnt. Double-buffered x tiles: 2 x 16,512 B.
  __shared__ float xbuf[2][kRows * kStride];

  const int tid   = threadIdx.x;                // one lane per batch row
  const int row_g = blockIdx.x * kRows + tid;

  // hidden=1: weights are scalars. Pre-halve i/f/o gate params so that
  // sigmoid(p) = 0.5*tanh(0.5*p)+0.5 needs no extra multiply on the
  // h-dependent critical path.
  const float wi0 = 0.5f * w_ih[0], wi1 = 0.5f * w_ih[1];
  const float wi2 =        w_ih[2], wi3 = 0.5f * w_ih[3];
  const float wh0 = 0.5f * w_hh[0], wh1 = 0.5f * w_hh[1];
  const float wh2 =        w_hh[2], wh3 = 0.5f * w_hh[3];
  const float b0 = 0.5f * (b_ih[0] + b_hh[0]);
  const float b1 = 0.5f * (b_ih[1] + b_hh[1]);
  const float b2 =        (b_ih[2] + b_hh[2]);
  const float b3 = 0.5f * (b_ih[3] + b_hh[3]);

  const u64 xblk = (u64)(const char*)x
                 + (u64)blockIdx.x * kRows * kT * sizeof(float);

  // Prologue: DMA tile 0 into buffer 0.
  tdm_load_tile(xblk, lds_off(&xbuf[0][0]));

  float h = 0.0f, c = 0.0f;
  float* __restrict__ outp = out + (u64)row_g * kT;

  for (int ti = 0; ti < kTiles; ++ti) {
    const int buf = ti & 1;

    // Kick off the next tile's DMA, then wait until only that newest op is
    // still outstanding (TDM completes in order per wave) -> tile ti is
    // resident while tile ti+1 streams in behind the compute.
    if (ti + 1 < kTiles) {
      tdm_load_tile(xblk + (u64)(ti + 1) * kTileT * sizeof(float),
                    lds_off(&xbuf[buf ^ 1][0]));
      __builtin_amdgcn_s_wait_tensorcnt(1);
    } else {
      __builtin_amdgcn_s_wait_tensorcnt(0);
    }

    const float* __restrict__ rowp = &xbuf[buf][tid * kStride];
    const int t0 = ti * kTileT;

    for (int cc = 0; cc < kTileT; cc += 4) {
      float hr[4];
#pragma unroll
      for (int k = 0; k < 4; ++k) {
        const float xt = rowp[cc + k];     // bank-conflict-free ds_load_b32
        // x-dependent halves: independent of h, schedule ahead of the chain.
        const float px0 = fmaf(xt, wi0, b0);
        const float px1 = fmaf(xt, wi1, b1);
        const float px2 = fmaf(xt, wi2, b2);
        const float px3 = fmaf(xt, wi3, b3);
        const float ai = fmaf(h, wh0, px0);        // 0.5 * preact(i)
        const float af = fmaf(h, wh1, px1);        // 0.5 * preact(f)
        const float ag = fmaf(h, wh2, px2);        //       preact(g)
        const float ao = fmaf(h, wh3, px3);        // 0.5 * preact(o)
        const float ig = fmaf(0.5f, FAST_TANH(ai), 0.5f);
        const float fg = fmaf(0.5f, FAST_TANH(af), 0.5f);
        const float og = fmaf(0.5f, FAST_TANH(ao), 0.5f);
        const float gg = FAST_TANH(ag);
        c = fmaf(fg, c, ig * gg);
        h = og * FAST_TANH(c);
        hr[k] = h;
      }
      // 16B aligned -> one global_store_b128 per 4 timesteps; partial lines
      // merge in the 192MB L2 before HBM writeback.
      *(float4*)(outp + t0 + cc) = make_float4(hr[0], hr[1], hr[2], hr[3]);
    }
  }
}

extern "C" void kernel_launch(void* const* d_in, const int* in_sizes, int n_in,
                              void* d_out, int out_size, void* d_ws, size_t ws_size,
                              hipStream_t stream) {
  (void)n_in; (void)out_size; (void)d_ws; (void)ws_size;
  const float* x    = (const float*)d_in[0];
  const float* w_ih = (const float*)d_in[1];
  const float* w_hh = (const float*)d_in[2];
  const float* b_ih = (const float*)d_in[3];
  const float* b_hh = (const float*)d_in[4];
  float* out = (float*)d_out;

  const int B = in_sizes[0] / kT;               // 16384
  dim3 grid(B / kRows), block(kRows);           // 512 single-wave blocks
  hipLaunchKernelGGL(OneCell_lstm_kernel, grid, block, 0, stream,
                     x, w_ih, w_hh, b_ih, b_hh, out);
}